// MultiHeadAttention_71339406787396
// MI455X (gfx1250) — compile-verified
//
#include <hip/hip_runtime.h>
#include <hip/hip_bf16.h>

// Problem dims (fixed by the reference)
#define B_ 4
#define N_ 256
#define D_ 256
#define H_ 8
#define DK_ 32
#define DV_ 32
#define R_ 64
#define BN_ (B_ * N_)          // 1024
#define SCALE_QK 0.17677669529663687f  // 1/sqrt(32)

typedef __attribute__((ext_vector_type(2))) float v2f;
typedef __attribute__((ext_vector_type(8))) float v8f;

static __device__ __forceinline__ v8f wmma_f32(v2f a, v2f b, v8f c) {
  // V_WMMA_F32_16X16X4_F32: D = A(16x4) * B(4x16) + C(16x16), fp32
  return __builtin_amdgcn_wmma_f32_16x16x4_f32(false, a, false, b, (short)0, c,
                                               false, false);
}

// ---------------------------------------------------------------------------
// Generic batched GEMM, C = A @ B (row-major), one wave per 16x16 C tile.
// Batch index z decomposes as (bz, hz) with z = bz*batchH + hz; per-batch
// element offsets are bz*s?b + hz*s?h.  All M,N,K are multiples of 16/4 here.
// ---------------------------------------------------------------------------
__global__ void gemm_nn_ker(const float* __restrict__ A,
                            const float* __restrict__ B, float* __restrict__ C,
                            int M, int N, int K, int lda, int ldb, int ldc,
                            int batchH, long sAb, long sAh, long sBb, long sBh,
                            long sCb, long sCh, int accumulate) {
  int lane = threadIdx.x & 31;
  int lh   = lane >> 4;        // half-wave (selects K pair / M+8)
  int mrow = lane & 15;
  int tN = blockIdx.x * 16;
  int tM = blockIdx.y * 16;
  int z = blockIdx.z;
  int bz = z / batchH, hz = z % batchH;
  const float* Ab = A + bz * sAb + hz * sAh;
  const float* Bb = B + bz * sBb + hz * sBh;
  float* Cb = C + bz * sCb + hz * sCh;

  v8f acc = {0.f, 0.f, 0.f, 0.f, 0.f, 0.f, 0.f, 0.f};
  if (accumulate) {
#pragma unroll
    for (int r = 0; r < 8; ++r)
      acc[r] = Cb[(tM + r + 8 * lh) * ldc + tN + mrow];
  }
  const float* arow = Ab + (long)(tM + mrow) * lda;
  for (int k0 = 0; k0 < K; k0 += 4) {
    int ka = k0 + 2 * lh;
    v2f a; a.x = arow[ka]; a.y = arow[ka + 1];
    v2f bfr; bfr.x = Bb[(long)ka * ldb + tN + mrow];
    bfr.y = Bb[(long)(ka + 1) * ldb + tN + mrow];
    acc = wmma_f32(a, bfr, acc);
  }
#pragma unroll
  for (int r = 0; r < 8; ++r)
    Cb[(tM + r + 8 * lh) * ldc + tN + mrow] = acc[r];
}

// C = A @ B^T : B stored row-major [N,K]; B fragment loads mirror A's.
__global__ void gemm_nt_ker(const float* __restrict__ A,
                            const float* __restrict__ B, float* __restrict__ C,
                            int M, int N, int K, int lda, int ldb, int ldc,
                            int batchH, long sAb, long sAh, long sBb, long sBh,
                            long sCb, long sCh) {
  int lane = threadIdx.x & 31;
  int lh   = lane >> 4;
  int mrow = lane & 15;
  int tN = blockIdx.x * 16;
  int tM = blockIdx.y * 16;
  int z = blockIdx.z;
  int bz = z / batchH, hz = z % batchH;
  const float* Ab = A + bz * sAb + hz * sAh;
  const float* Bb = B + bz * sBb + hz * sBh;
  float* Cb = C + bz * sCb + hz * sCh;

  v8f acc = {0.f, 0.f, 0.f, 0.f, 0.f, 0.f, 0.f, 0.f};
  const float* arow = Ab + (long)(tM + mrow) * lda;
  const float* brow = Bb + (long)(tN + mrow) * ldb;
  for (int k0 = 0; k0 < K; k0 += 4) {
    int ka = k0 + 2 * lh;
    v2f a;  a.x = arow[ka];  a.y = arow[ka + 1];
    v2f bf; bf.x = brow[ka]; bf.y = brow[ka + 1];
    acc = wmma_f32(a, bf, acc);
  }
#pragma unroll
  for (int r = 0; r < 8; ++r)
    Cb[(tM + r + 8 * lh) * ldc + tN + mrow] = acc[r];
}

// ---------------------------------------------------------------------------
// Fused per-(b,i) kernel: stages link[b,i,:,:] (256x64 f32 = 64KB) in LDS once
// (via the gfx1250 async Global->LDS path when available), then:
//   bias = c @ link^T (WMMA, M padded 8->16) -> scale+mask+softmax (alpha out)
//   w = alpha @ link (WMMA).
// Saves re-reading 67MB of link from HBM.  Block = 256 threads = 8 waves.
// ---------------------------------------------------------------------------
__global__ void fused_attn_ker(const float* __restrict__ link,
                               const float* __restrict__ cBuf,
                               const int* __restrict__ mask,
                               float* __restrict__ alphaOut,  // holds S on entry
                               float* __restrict__ wBuf) {
  extern __shared__ float smem[];
  float* link_s = smem;           // [256][64]
  float* sc_s   = smem + 16384;   // [8][256]  scores -> alpha
  float* c_s    = smem + 18432;   // [8][64]

  const int bi = blockIdx.x;          // b*256 + i
  const int b = bi >> 8, i = bi & 255;
  const int tid = threadIdx.x;
  const int lane = tid & 31, wid = tid >> 5;
  const int lh = lane >> 4, mrow = lane & 15;

  // ---- stage link tile: async Global->LDS (ASYNCcnt) if available ---------
  const float* lsrcf = link + (long)bi * (N_ * R_);
#if __has_builtin(__builtin_amdgcn_global_load_async_to_lds_b128)
  {
    typedef int async_v4i __attribute__((vector_size(16)));
    typedef __attribute__((address_space(1))) async_v4i* gp_t;
    typedef __attribute__((address_space(3))) async_v4i* lp_t;
    // 4096 x b128 transfers, 16 per thread; LDS written directly (no VGPRs)
    for (int t = tid; t < (N_ * R_) / 4; t += 256) {
      gp_t g = (gp_t)(uintptr_t)(lsrcf + t * 4);
      lp_t l = (lp_t)(uintptr_t)(link_s + t * 4);
      __builtin_amdgcn_global_load_async_to_lds_b128(g, l, 0, 0);
    }
  }
#else
  {
    const float4* lsrc = (const float4*)lsrcf;
    float4* ldst = (float4*)link_s;
    for (int t = tid; t < (N_ * R_) / 4; t += 256) ldst[t] = lsrc[t];
  }
#endif
  // stage c[b,:,i,:]  (8x64)  -- overlaps with async link DMA
  for (int t = tid; t < H_ * R_; t += 256) {
    int h = t >> 6, r = t & 63;
    c_s[t] = cBuf[((long)(b * H_ + h) * N_ + i) * R_ + r];
  }
  // stage S rows (8x256) from alpha region
  for (int t = tid; t < H_ * N_; t += 256) {
    int h = t >> 8, j = t & 255;
    sc_s[t] = alphaOut[((long)(b * H_ + h) * N_ + i) * N_ + j];
  }
#if __has_builtin(__builtin_amdgcn_global_load_async_to_lds_b128)
#if __has_builtin(__builtin_amdgcn_s_wait_asynccnt)
  __builtin_amdgcn_s_wait_asynccnt(0);
#else
  asm volatile("s_wait_asynccnt 0x0" ::: "memory");
#endif
#endif
  __syncthreads();

  // --- bias = c(8x64,padded to 16) @ link^T, combined with scale + mask -----
  const int* mrowp = mask + (long)bi * N_;
#pragma unroll
  for (int t = 0; t < 2; ++t) {
    int tN = (wid * 2 + t) * 16;
    v8f acc = {0.f, 0.f, 0.f, 0.f, 0.f, 0.f, 0.f, 0.f};
    for (int k0 = 0; k0 < R_; k0 += 4) {
      int ka = k0 + 2 * lh;
      v2f a;
      a.x = (mrow < 8) ? c_s[mrow * R_ + ka] : 0.f;
      a.y = (mrow < 8) ? c_s[mrow * R_ + ka + 1] : 0.f;
      v2f bf;                                 // B^T: rows are j
      bf.x = link_s[(tN + mrow) * R_ + ka];
      bf.y = link_s[(tN + mrow) * R_ + ka + 1];
      acc = wmma_f32(a, bf, acc);
    }
    if (lh == 0) {                            // rows 0..7 live in lanes 0-15
      int col = tN + mrow;
      int mv = mrowp[col];
#pragma unroll
      for (int r = 0; r < 8; ++r) {
        float v = (sc_s[r * N_ + col] + acc[r]) * SCALE_QK;
        sc_s[r * N_ + col] = (mv == 0) ? -1e9f : v;
      }
    }
  }
  __syncthreads();

  // --- softmax over j for head h = wid (8 elems / lane) ---------------------
  {
    int h = wid;
    float vals[8], mx = -3.4e38f;
#pragma unroll
    for (int e = 0; e < 8; ++e) {
      vals[e] = sc_s[h * N_ + lane + e * 32];
      mx = fmaxf(mx, vals[e]);
    }
#pragma unroll
    for (int off = 16; off >= 1; off >>= 1) mx = fmaxf(mx, __shfl_xor(mx, off, 32));
    float sum = 0.f;
#pragma unroll
    for (int e = 0; e < 8; ++e) { vals[e] = __expf(vals[e] - mx); sum += vals[e]; }
#pragma unroll
    for (int off = 16; off >= 1; off >>= 1) sum += __shfl_xor(sum, off, 32);
    float inv = 1.f / sum;
    float* arow = alphaOut + ((long)(b * H_ + h) * N_ + i) * N_;
#pragma unroll
    for (int e = 0; e < 8; ++e) {
      float al = vals[e] * inv;
      sc_s[h * N_ + lane + e * 32] = al;
      arow[lane + e * 32] = al;
    }
  }
  __syncthreads();

  // --- w = alpha(8x256,padded) @ link(256x64): waves 0..3, one 16-col tile --
  if (wid < 4) {
    int tN = wid * 16;
    v8f acc = {0.f, 0.f, 0.f, 0.f, 0.f, 0.f, 0.f, 0.f};
    for (int k0 = 0; k0 < N_; k0 += 4) {
      int ka = k0 + 2 * lh;
      v2f a;
      a.x = (mrow < 8) ? sc_s[mrow * N_ + ka] : 0.f;
      a.y = (mrow < 8) ? sc_s[mrow * N_ + ka + 1] : 0.f;
      v2f bf;
      bf.x = link_s[ka * R_ + tN + mrow];
      bf.y = link_s[(ka + 1) * R_ + tN + mrow];
      acc = wmma_f32(a, bf, acc);
    }
    if (lh == 0) {
#pragma unroll
      for (int r = 0; r < 8; ++r)       // w layout [b,i,h,r]
        wBuf[((long)bi * H_ + r) * R_ + tN + mrow] = acc[r];
    }
  }
}

// ---------------------------------------------------------------------------
// Residual + LayerNorm epilogue: one block per row of 256.
// ---------------------------------------------------------------------------
__global__ void ln_ker(const float* __restrict__ outlin,
                       const float* __restrict__ resid,
                       const float* __restrict__ gamma,
                       const float* __restrict__ beta, float* __restrict__ out) {
  __shared__ float red1[8], red2[8];
  int row = blockIdx.x, tid = threadIdx.x;
  int lane = tid & 31, wid = tid >> 5;
  long idx = (long)row * D_ + tid;
  float v = outlin[idx] + resid[idx];
  float s = v;
#pragma unroll
  for (int off = 16; off >= 1; off >>= 1) s += __shfl_xor(s, off, 32);
  if (lane == 0) red1[wid] = s;
  __syncthreads();
  float tot = 0.f;
#pragma unroll
  for (int k = 0; k < 8; ++k) tot += red1[k];
  float mu = tot * (1.f / D_);
  float d = v - mu;
  float s2 = d * d;
#pragma unroll
  for (int off = 16; off >= 1; off >>= 1) s2 += __shfl_xor(s2, off, 32);
  if (lane == 0) red2[wid] = s2;
  __syncthreads();
  float tot2 = 0.f;
#pragma unroll
  for (int k = 0; k < 8; ++k) tot2 += red2[k];
  float var = tot2 * (1.f / D_);
  out[idx] = d * rsqrtf(var + 1e-6f) * gamma[tid] + beta[tid];
}

// ---------------------------------------------------------------------------
extern "C" void kernel_launch(void* const* d_in, const int* in_sizes, int n_in,
                              void* d_out, int out_size, void* d_ws,
                              size_t ws_size, hipStream_t stream) {
  const float* q    = (const float*)d_in[0];
  // d_in[1] (k) and d_in[2] (v) are unused in this reference branch.
  const int*   mask = (const int*)d_in[3];
  const float* link = (const float*)d_in[4];
  const float* Wq   = (const float*)d_in[5];
  const float* Wk   = (const float*)d_in[6];
  const float* Wr   = (const float*)d_in[7];
  const float* Wv   = (const float*)d_in[8];
  const float* Wvv  = (const float*)d_in[9];
  const float* relE = (const float*)d_in[10];
  const float* Wo   = (const float*)d_in[11];
  const float* gamma = (const float*)d_in[12];
  const float* beta  = (const float*)d_in[13];

  float* out    = (float*)d_out;                 // [4,256,256]
  float* alphaR = out + (long)BN_ * D_;          // [4,8,256,256] (S first, then alpha)

  // workspace layout (floats)
  float* ws = (float*)d_ws;
  float* XqW   = ws;                 // 1024*256
  float* XkW   = XqW + 262144;
  float* XvW   = XkW + 262144;
  float* Er    = XvW + 262144;       // 64*256
  float* Ev    = Er + 16384;
  float* cBuf  = Ev + 16384;         // [b,h,n,r] 524288
  float* wBuf  = cBuf + 524288;      // [b,n,h,r] 524288
  float* Zc    = wBuf + 524288;      // [b,n,h*dv] 262144
  float* outlin = Zc + 262144;       // 262144

  dim3 blk(32);

  // Projections: [1024,256] @ [256,256]
  gemm_nn_ker<<<dim3(16, 64, 1), blk, 0, stream>>>(q, Wq, XqW, 1024, 256, 256,
      256, 256, 256, 1, 0, 0, 0, 0, 0, 0, 0);
  gemm_nn_ker<<<dim3(16, 64, 1), blk, 0, stream>>>(q, Wk, XkW, 1024, 256, 256,
      256, 256, 256, 1, 0, 0, 0, 0, 0, 0, 0);
  gemm_nn_ker<<<dim3(16, 64, 1), blk, 0, stream>>>(q, Wv, XvW, 1024, 256, 256,
      256, 256, 256, 1, 0, 0, 0, 0, 0, 0, 0);
  // Er = relationE @ Wr, Ev = relationE @ Wvv : [64,256]@[256,256]
  gemm_nn_ker<<<dim3(16, 4, 1), blk, 0, stream>>>(relE, Wr, Er, 64, 256, 256,
      256, 256, 256, 1, 0, 0, 0, 0, 0, 0, 0);
  gemm_nn_ker<<<dim3(16, 4, 1), blk, 0, stream>>>(relE, Wvv, Ev, 64, 256, 256,
      256, 256, 256, 1, 0, 0, 0, 0, 0, 0, 0);

  // S[b,h] = Xq_h @ Xk_h^T -> into alpha region (overwritten by softmax later)
  gemm_nt_ker<<<dim3(16, 16, 32), blk, 0, stream>>>(XqW, XkW, alphaR, 256, 256,
      32, 256, 256, 256, 8, 65536, 32, 65536, 32, 524288, 65536);
  // c[b,h,i,r] = Xq_h @ Er_h^T : M=256,N=64,K=32
  gemm_nt_ker<<<dim3(4, 16, 32), blk, 0, stream>>>(XqW, Er, cBuf, 256, 64, 32,
      256, 256, 64, 8, 65536, 32, 0, 32, 131072, 16384);

  // Fused bias + softmax + w, link staged once in 76KB dynamic LDS
  fused_attn_ker<<<dim3(BN_), dim3(256), 75776, stream>>>(link, cBuf, mask,
                                                          alphaR, wBuf);

  // Z1[b,h] = alpha @ Xv_h -> Zc[b,n,h*32+d]
  gemm_nn_ker<<<dim3(2, 16, 32), blk, 0, stream>>>(alphaR, XvW, Zc, 256, 32,
      256, 256, 256, 256, 8, 524288, 65536, 65536, 32, 65536, 32, 0);
  // Z2 += w_h @ Ev_h : M=1024,N=32,K=64, accumulate into Zc
  gemm_nn_ker<<<dim3(2, 64, 8), blk, 0, stream>>>(wBuf, Ev, Zc, 1024, 32, 64,
      512, 256, 256, 8, 0, 64, 0, 32, 0, 32, 1);
  // outlin = Zc @ Wo
  gemm_nn_ker<<<dim3(16, 64, 1), blk, 0, stream>>>(Zc, Wo, outlin, 1024, 256,
      256, 256, 256, 256, 1, 0, 0, 0, 0, 0, 0, 0);

  // out = LayerNorm(outlin + q) * gamma + beta
  ln_ker<<<dim3(BN_), dim3(256), 0, stream>>>(outlin, q, gamma, beta, out);
}